// HIE_10780367913780
// MI455X (gfx1250) — compile-verified
//
#include <hip/hip_runtime.h>
#include <hip/hip_bf16.h>
#include <stdint.h>

typedef float v2f __attribute__((ext_vector_type(2)));
typedef float v8f __attribute__((ext_vector_type(8)));
typedef unsigned int u32x4 __attribute__((ext_vector_type(4)));
typedef int i32x4 __attribute__((ext_vector_type(4)));
typedef int i32x8 __attribute__((ext_vector_type(8)));

#define NNODES 50000
#define DIM 128
#define LAYERS 3
#define RANK 16
#define WPITCH (DIM + 4)   /* 132-dword pitch: ds_load_b64 hits all 64 banks once */
#define CPITCH (DIM + 8)   /* 136-dword pitch for W_col */

#if defined(__has_builtin)
#if __has_builtin(__builtin_amdgcn_tensor_load_to_lds)
#define HAVE_TDM 1
#endif
#endif
#if defined(__has_include)
#if __has_include(<hip/amd_detail/amd_gfx1250_TDM.h>)
#define TDM_6ARG 1   /* therock headers => 6-arg builtin arity */
#endif
#endif

__device__ __forceinline__ void tdm_wait0() {
#if defined(__has_builtin) && __has_builtin(__builtin_amdgcn_s_wait_tensorcnt)
  __builtin_amdgcn_s_wait_tensorcnt(0);
#else
  asm volatile("s_wait_tensorcnt 0x0" ::: "memory");
#endif
}

#if defined(HAVE_TDM)
// 2D TDM load: rows x width fp32 tile (row stride = stride elems) from global
// into LDS at lds_off, inserting `pad_code+1` dwords of padding every 128
// dwords (pad_interval code 6) to build a bank-conflict-free pitch.
__device__ __forceinline__ void tdm_load_2d(const void* gptr, uint32_t lds_off,
                                            int width, int rows, int stride,
                                            int pad_code) {
  uint64_t ga = (uint64_t)gptr;
  u32x4 g0;
  g0.x = 1u;                                        // count=1, user descriptor
  g0.y = lds_off;                                   // lds_addr (bytes)
  g0.z = (uint32_t)(ga & 0xFFFFFFFFu);              // global_addr[31:0]
  g0.w = (uint32_t)((ga >> 32) & 0x1FFFFFFu) | (2u << 30);  // addr[56:32]|type=2
  i32x8 g1;
  int d0 = (2 << 16);                               // data_size = 4B
  if (pad_code >= 0)
    d0 |= (1 << 20) | (6 << 22) | (pad_code << 25); // pad_en, every 128 dwords
  g1[0] = d0;
  g1[1] = (width & 0xFFFF) << 16;                   // tensor_dim0 lo16
  g1[2] = ((width >> 16) & 0xFFFF) | ((rows & 0xFFFF) << 16);  // dim0 hi | dim1 lo
  g1[3] = ((rows >> 16) & 0xFFFF) | ((width & 0xFFFF) << 16);  // dim1 hi | tile_dim0
  g1[4] = (rows & 0xFFFF);                          // tile_dim1 (tile_dim2 = 0)
  g1[5] = stride;                                   // tensor_dim0_stride lo32
  g1[6] = 0;
  g1[7] = 0;
  i32x4 z4 = {0, 0, 0, 0};
#if defined(TDM_6ARG)
  i32x8 z8 = {0, 0, 0, 0, 0, 0, 0, 0};
  __builtin_amdgcn_tensor_load_to_lds(g0, g1, z4, z4, z8, 0);
#else
  __builtin_amdgcn_tensor_load_to_lds(g0, g1, z4, z4, 0);
#endif
}
#endif

// ---------------- init: h = h_sum = node_feats ----------------
__global__ void k_init(const float4* __restrict__ src, float4* __restrict__ h,
                       float4* __restrict__ hsum, int n4) {
  int i = blockIdx.x * blockDim.x + threadIdx.x;
  if (i < n4) { float4 v = src[i]; h[i] = v; hsum[i] = v; }
}

// ---------------- SpMM scatter: agg[row] += val * h[col] ----------------
__global__ void k_spmm(const int* __restrict__ rows, const int* __restrict__ cols,
                       const float* __restrict__ vals, const float* __restrict__ h,
                       float* __restrict__ agg, int E) {
  int lane = threadIdx.x & 31;
  int wave = blockIdx.x * (blockDim.x >> 5) + (threadIdx.x >> 5);
  int nw   = gridDim.x * (blockDim.x >> 5);
  for (int e = wave; e < E; e += nw) {
    int r = rows[e], c = cols[e];
    float v = vals[e];
    float4 hv = ((const float4*)(h + (size_t)c * DIM))[lane];
    float* dst = agg + (size_t)r * DIM + lane * 4;
    atomicAdd(dst + 0, v * hv.x);
    atomicAdd(dst + 1, v * hv.y);
    atomicAdd(dst + 2, v * hv.z);
    atomicAdd(dst + 3, v * hv.w);
  }
}

// ------ GEMM + bias + ELU: W staged in LDS (TDM), A double-buffered --------
// out[n][d] = elu( sum_k A[n][k]*W[d][k] + b[d] );  h = out; hsum += out.
__global__ void k_gemm_elu(const float* __restrict__ A, const float* __restrict__ W,
                           const float* __restrict__ bias, float* __restrict__ h,
                           float* __restrict__ hsum, int ntiles) {
  extern __shared__ float Wl[];   // DIM rows x WPITCH dwords (padded, 67.6 KB)
  int tid = threadIdx.x;
#if defined(HAVE_TDM)
  if ((tid >> 5) == 0) {          // one wave issues the DMA descriptor
    tdm_load_2d(W, (uint32_t)(uintptr_t)Wl, DIM, DIM, DIM, /*+4 dwords*/ 3);
    tdm_wait0();
  }
#else
  for (int i = tid; i < DIM * DIM / 4; i += blockDim.x) {
    int row = i >> 5, k4 = i & 31;
    *(float4*)(&Wl[row * WPITCH + k4 * 4]) = ((const float4*)W)[i];
  }
#endif
  __syncthreads();

  int wave = blockIdx.x * (blockDim.x >> 5) + (tid >> 5);
  if (wave >= ntiles) return;            // whole-wave exit: EXEC stays all-1s
  int lane = tid & 31;
  int lm = lane & 15, hi = lane >> 4;
  int m0 = wave * 16;

  const float* Ar = A + (size_t)(m0 + lm) * DIM + 2 * hi;   // 8B-aligned base
  const float* Bbase = &Wl[lm * WPITCH + 2 * hi];           // + j*16*WPITCH + k0

  v8f acc[8] = {};
  v2f cur[8], nxt[8];
#pragma unroll
  for (int t = 0; t < 8; ++t)
    cur[t] = *(const v2f*)__builtin_assume_aligned(Ar + t * 4, 8);

#pragma unroll 1                          // keep chunks in flight, not hoisted
  for (int c = 0; c < 4; ++c) {
    if (c < 3) {
#pragma unroll
      for (int t = 0; t < 8; ++t)
        nxt[t] = *(const v2f*)__builtin_assume_aligned(Ar + (c + 1) * 32 + t * 4, 8);
    }
    const float* Bc = Bbase + c * 32;
#pragma unroll
    for (int t = 0; t < 8; ++t) {
#pragma unroll
      for (int j = 0; j < 8; ++j) {
        // ds_load_b64; j*16*WPITCH*4 + t*16 folds into the 16-bit offset field
        v2f b = *(const v2f*)__builtin_assume_aligned(
            Bc + j * 16 * WPITCH + t * 4, 8);
        acc[j] = __builtin_amdgcn_wmma_f32_16x16x4_f32(
            false, cur[t], false, b, (short)0, acc[j], false, false);
      }
    }
#pragma unroll
    for (int t = 0; t < 8; ++t) cur[t] = nxt[t];
  }

#pragma unroll
  for (int j = 0; j < 8; ++j) {
    int col = j * 16 + lm;
    float bb = bias[col];
#pragma unroll
    for (int r = 0; r < 8; ++r) {
      int row = m0 + r + 8 * hi;
      float x = acc[j][r] + bb;
      float y = x > 0.0f ? x : (__expf(x) - 1.0f);   // ELU
      size_t idx = (size_t)row * DIM + col;
      h[idx] = y;
      hsum[idx] += y;
    }
  }
}

// ---------------- column sum of hsum -> z ----------------
__global__ void k_colsum(const float* __restrict__ hsum, float* __restrict__ z, int n) {
  int c = threadIdx.x & 127;
  int r0 = blockIdx.x * 2 + (threadIdx.x >> 7);
  int stride = gridDim.x * 2;
  float acc = 0.0f;
  for (int r = r0; r < n; r += stride) acc += hsum[(size_t)r * DIM + c];
  atomicAdd(&z[c], acc);
}

// ---------------- hz = leaky_relu(z @ fc1_W^T + fc1_b) ----------------
__global__ void k_hz(const float* __restrict__ zsum, const float* __restrict__ W,
                     const float* __restrict__ b, float* __restrict__ hz, float scale) {
  __shared__ float zs[DIM];
  int t = threadIdx.x;
  zs[t] = zsum[t] * scale;
  __syncthreads();
  float acc = b[t];
  for (int k = 0; k < DIM; ++k) acc += zs[k] * W[t * DIM + k];
  hz[t] = acc > 0.0f ? acc : 0.2f * acc;
}

// ---------------- out[r] = M[r,:] . v + b[r] (row per wave-iter) ----------
__global__ void k_matvec(const float* __restrict__ M, const float* __restrict__ v,
                         const float* __restrict__ b, float* __restrict__ out, int rows) {
  int lane = threadIdx.x & 31;
  int wave = blockIdx.x * (blockDim.x >> 5) + (threadIdx.x >> 5);
  int nw   = gridDim.x * (blockDim.x >> 5);
  float4 hv = ((const float4*)v)[lane];
  for (int r = wave; r < rows; r += nw) {
    float4 w = ((const float4*)(M + (size_t)r * DIM))[lane];   // 512B coalesced
    float p = w.x * hv.x + w.y * hv.y + w.z * hv.z + w.w * hv.w;
#pragma unroll
    for (int off = 16; off > 0; off >>= 1) p += __shfl_xor(p, off, 32);
    if (lane == 0) out[r] = p + b[r];
  }
}

// ------- out = ini * (1 + W_row @ W_col), K=16, W_col staged in LDS -------
__global__ void k_lowrank(const float* __restrict__ Wrow, const float* __restrict__ Wcol,
                          const float* __restrict__ ini, float* __restrict__ out,
                          int ntiles) {
  extern __shared__ float Cl[];   // RANK rows x CPITCH dwords (8.7 KB)
  int tid = threadIdx.x;
#if defined(HAVE_TDM)
  if ((tid >> 5) == 0) {
    tdm_load_2d(Wcol, (uint32_t)(uintptr_t)Cl, DIM, RANK, DIM, /*+8 dwords*/ 7);
    tdm_wait0();
  }
#else
  for (int i = tid; i < RANK * DIM / 4; i += blockDim.x) {
    int r = i >> 5, k4 = i & 31;
    *(float4*)(&Cl[r * CPITCH + k4 * 4]) = ((const float4*)Wcol)[i];
  }
#endif
  __syncthreads();

  int wave = blockIdx.x * (blockDim.x >> 5) + (tid >> 5);
  if (wave >= ntiles) return;
  int lane = tid & 31;
  int lm = lane & 15, hi = lane >> 4;
  int m0 = wave * 16;

  const float* Ar = Wrow + (size_t)(m0 + lm) * RANK + 2 * hi;
  v2f af[4];
#pragma unroll
  for (int t = 0; t < 4; ++t)
    af[t] = *(const v2f*)__builtin_assume_aligned(Ar + t * 4, 8);

  v8f acc[8] = {};
#pragma unroll
  for (int t = 0; t < 4; ++t) {
    int k0 = t * 4;
#pragma unroll
    for (int j = 0; j < 8; ++j) {
      int col = j * 16 + lm;
      v2f b;
      b.x = Cl[(k0 + 2 * hi) * CPITCH + col];      // B[k][n] = W_col[k][n]
      b.y = Cl[(k0 + 2 * hi + 1) * CPITCH + col];
      acc[j] = __builtin_amdgcn_wmma_f32_16x16x4_f32(
          false, af[t], false, b, (short)0, acc[j], false, false);
    }
  }
#pragma unroll
  for (int j = 0; j < 8; ++j) {
    int col = j * 16 + lm;
#pragma unroll
    for (int r = 0; r < 8; ++r) {
      int row = m0 + r + 8 * hi;
      size_t idx = (size_t)row * DIM + col;
      float e = ini[idx];
      out[idx] = e + acc[j][r] * e;
    }
  }
}

extern "C" void kernel_launch(void* const* d_in, const int* in_sizes, int n_in,
                              void* d_out, int out_size, void* d_ws, size_t ws_size,
                              hipStream_t stream) {
  const int*   adj_rows   = (const int*)d_in[0];
  const int*   adj_cols   = (const int*)d_in[1];
  const float* adj_vals   = (const float*)d_in[2];
  const float* node_feats = (const float*)d_in[3];
  const float* gnn_W      = (const float*)d_in[4];
  const float* gnn_b      = (const float*)d_in[5];
  const float* fc1_W      = (const float*)d_in[6];
  const float* fc1_b      = (const float*)d_in[7];
  const float* fcrow_W    = (const float*)d_in[8];
  const float* fcrow_b    = (const float*)d_in[9];
  const float* fccol_W    = (const float*)d_in[10];
  const float* fccol_b    = (const float*)d_in[11];
  const float* ini        = (const float*)d_in[12];
  float* out = (float*)d_out;

  const int E = in_sizes[0];
  const int N = NNODES;
  const size_t nd = (size_t)N * DIM;

  // workspace carve-out (fp32): h | hsum | agg | Wrow | Wcol | z | hz
  float* ws   = (float*)d_ws;
  float* h    = ws;  ws += nd;
  float* hsum = ws;  ws += nd;
  float* agg  = ws;  ws += nd;
  float* Wrow = ws;  ws += (size_t)N * RANK;
  float* Wcol = ws;  ws += (size_t)RANK * DIM;
  float* z    = ws;  ws += DIM;
  float* hz   = ws;  ws += DIM;

  int n4 = (int)(nd / 4);
  k_init<<<(n4 + 255) / 256, 256, 0, stream>>>(
      (const float4*)node_feats, (float4*)h, (float4*)hsum, n4);

  const int ntiles = N / 16;   // 3125 (exact)
  const size_t gemm_lds = (size_t)DIM * WPITCH * sizeof(float);    // 67,584 B
  const size_t lr_lds   = (size_t)RANK * CPITCH * sizeof(float);   //  8,704 B
  for (int l = 0; l < LAYERS; ++l) {
    hipMemsetAsync(agg, 0, nd * sizeof(float), stream);
    k_spmm<<<2048, 256, 0, stream>>>(adj_rows, adj_cols, adj_vals, h, agg, E);
    k_gemm_elu<<<(ntiles + 7) / 8, 256, gemm_lds, stream>>>(
        agg, gnn_W + (size_t)l * DIM * DIM, gnn_b + (size_t)l * DIM,
        h, hsum, ntiles);
  }

  hipMemsetAsync(z, 0, DIM * sizeof(float), stream);
  k_colsum<<<512, 256, 0, stream>>>(hsum, z, N);
  k_hz<<<1, DIM, 0, stream>>>(z, fc1_W, fc1_b, hz,
                              1.0f / ((float)(LAYERS + 1) * (float)N));
  k_matvec<<<1024, 256, 0, stream>>>(fcrow_W, hz, fcrow_b, Wrow, N * RANK);
  k_matvec<<<16,   256, 0, stream>>>(fccol_W, hz, fccol_b, Wcol, RANK * DIM);
  k_lowrank<<<(ntiles + 7) / 8, 256, lr_lds, stream>>>(Wrow, Wcol, ini, out, ntiles);
}